// SphericalBasisLayer_46024869543995
// MI455X (gfx1250) — compile-verified
//
#include <hip/hip_runtime.h>
#include <math.h>

// ---------------------------------------------------------------------------
// SphericalBasisLayer for MI455X (gfx1250).
// Phase 1: per-edge rbf_env table (500k x 42 f32, 84MB) into d_ws (L2-resident).
// Phase 2: per-triplet gather * spherical harmonics. The Legendre/ynorm factor
//          is computed as a genuine 16x16x4 f32 WMMA matmul (powers x coeffs).
//          Output stream uses non-temporal stores so the table stays in L2.
// Fallback fused kernel if ws_size is too small for the table.
// ---------------------------------------------------------------------------

#define NUM_L 7
#define NUM_R 6
#define LR 42

typedef float v2f __attribute__((ext_vector_type(2)));
typedef float v8f __attribute__((ext_vector_type(8)));

struct BesselConsts {
    float z[LR];   // zeros of spherical bessel j_l  (l-major, [l*6+n])
    float bn[LR];  // 1/sqrt(0.5*j_{l+1}(z)^2)
};

// ynorm[l] = sqrt((2l+1)/(4*pi))  (closed form, double-accurate literals)
__device__ __constant__ float YNORM[NUM_L] = {
    0.28209479177387814f, 0.4886025119029199f, 0.6307831305050401f,
    0.7463526651802308f,  0.8462843753216345f, 0.9356025796273889f,
    1.0171072362820548f
};

// BMAT[k][n] = ynorm[n] * (coefficient of x^k in Legendre P_n).  K padded to 8,
// N padded to 16 (WMMA B is 4x16 per instruction; columns 7..15 are zero).
__device__ __constant__ float BMAT[8][16] = {
  // k = 0
  { 0.28209479177387814f, 0.0f, -0.31539156525252005f, 0.0f,
    0.3173566407456129f,  0.0f, -0.3178460113381421f,  0.0f,0,0,0,0,0,0,0,0 },
  // k = 1
  { 0.0f, 0.4886025119029199f, 0.0f, -1.1195289977703462f,
    0.0f, 1.7542548368013542f, 0.0f, 0.0f,0,0,0,0,0,0,0,0 },
  // k = 2
  { 0.0f, 0.0f, 0.9461746957575601f, 0.0f,
    -3.1735664074561294f, 0.0f, 6.674766238100985f, 0.0f,0,0,0,0,0,0,0,0 },
  // k = 3
  { 0.0f, 0.0f, 0.0f, 1.865881662950577f,
    0.0f, -8.186522571739653f, 0.0f, 0.0f,0,0,0,0,0,0,0,0 },
  // k = 4
  { 0.0f, 0.0f, 0.0f, 0.0f,
    3.702494142032151f, 0.0f, -20.024298714302954f, 0.0f,0,0,0,0,0,0,0,0 },
  // k = 5
  { 0.0f, 0.0f, 0.0f, 0.0f, 0.0f, 7.3678703145656875f, 0.0f, 0.0f,
    0,0,0,0,0,0,0,0 },
  // k = 6
  { 0.0f, 0.0f, 0.0f, 0.0f, 0.0f, 0.0f, 14.684485723822166f, 0.0f,
    0,0,0,0,0,0,0,0 },
  // k = 7 (pad)
  { 0,0,0,0,0,0,0,0,0,0,0,0,0,0,0,0 }
};

// ---------------------------------------------------------------------------
// Device: evaluate the 42 rbf_env values for one edge, emitting each value.
// Mirrors the reference's float32 upward recurrence (same algorithm), with one
// reciprocal per element and a fused sincos.
// ---------------------------------------------------------------------------
template <typename EmitFn>
__device__ __forceinline__ void rbf_row(float d, float fenv,
                                        const BesselConsts& K, EmitFn emit) {
#pragma unroll
    for (int l = 0; l < NUM_L; ++l) {
#pragma unroll
        for (int n = 0; n < NUM_R; ++n) {
            const int i = l * NUM_R + n;
            float x = d * K.z[i];
            float s, c;
            sincosf(x, &s, &c);
            float rx = 1.0f / x;
            float j0 = s * rx;
            float j  = j0;
            if (l > 0) {
                float jm = j0;
                j = (j0 - c) * rx;   // == s/x^2 - c/x
#pragma unroll
                for (int kk = 1; kk < l; ++kk) {
                    float nx = (2.0f * kk + 1.0f) * rx * j - jm;
                    jm = j; j = nx;
                }
            }
            emit(l, i, fenv * (K.bn[i] * j));
        }
    }
}

__device__ __forceinline__ float envelope_scaled(float d) {
    // p = 6: env = 1/d - 28 d^5 + 48 d^6 - 21 d^7 ; u = (d<1)?env:0 ; * 5^-1.5
    float d2 = d * d, d4 = d2 * d2;
    float d5 = d4 * d, d6 = d4 * d2, d7 = d6 * d;
    float env = 1.0f / d + (-28.0f) * d5 + 48.0f * d6 + (-21.0f) * d7;
    float u = (d < 1.0f) ? env : 0.0f;
    return u * 0.08944271909999159f;   // (1/5)^1.5
}

// ---------------------------------------------------------------------------
// Phase 1: table[e][0..41] = rbf_env row of edge e  (kept RT -> L2 resident)
// ---------------------------------------------------------------------------
__global__ __launch_bounds__(256)
void sbl_edge_rbf_kernel(const float* __restrict__ D, float* __restrict__ table,
                         int nE, BesselConsts K) {
    int e = blockIdx.x * blockDim.x + threadIdx.x;
    if (e >= nE) return;
    float d = D[e] * 0.2f;
    float f = envelope_scaled(d);
    float* row = table + (size_t)e * LR;
    rbf_row(d, f, K, [&](int /*l*/, int i, float v) { row[i] = v; });
}

// ---------------------------------------------------------------------------
// Phase 2: one wave32 per 16-triplet tile.
//   sph[16 triplets][7] = WMMA( powers(16x8 f32) x BMAT(8x16 f32) )  -> LDS
//   then 21 fully-coalesced iterations write 16*42 outputs, gathering rows
//   from the L2-resident table.
// WMMA f32 16x16x4 operand layouts (ISA 7.12.2):
//   A (16x4): lane<16 -> {K0,K1}, lane>=16 -> {K2,K3}, M = lane&15
//   B (4x16): lane<16 -> rows {K0,K1}, lane>=16 -> rows {K2,K3}, N = lane&15
//   C (16x16): VGPR v, lane<16 -> M=v, lane>=16 -> M=v+8, N = lane&15
// ---------------------------------------------------------------------------
#define WPB 8   // waves per block

__global__ __launch_bounds__(256)
void sbl_triplet_wmma_kernel(const float* __restrict__ angle,
                             const int* __restrict__ id3,
                             const float* __restrict__ table,
                             float* __restrict__ out, int nT) {
    __shared__ float sph_s[WPB][16][8];   // [wave][triplet-in-tile][l] (pad 8)
    __shared__ int   id3_s[WPB][16];

    const int tid  = threadIdx.x;
    const int wv   = tid >> 5;
    const int lane = tid & 31;
    const int half = lane >> 4;
    const int m    = lane & 15;

    const int tile = blockIdx.x * WPB + wv;
    const int base = tile * 16;
    if (base >= nT) return;                 // wave-uniform: EXEC stays all-1s

    int t = base + m;
    if (t >= nT) t = nT - 1;                // clamp loads on the tail tile
    // angle/id3 are single-pass streams: non-temporal so the table keeps L2.
    const int g = __builtin_nontemporal_load(&id3[t]);

    // Prefetch the gathered row while we do the WMMA work (gfx1250 path).
    __builtin_prefetch(&table[(size_t)g * LR], 0, 3);

    const float ct = __cosf(__builtin_nontemporal_load(&angle[t]));
    const float c2 = ct * ct, c3 = c2 * ct, c4 = c2 * c2;
    const float c5 = c4 * ct, c6 = c4 * c2;

    // A: powers of ct. First WMMA covers K=0..3, second K=4..7 (K=7 row = 0).
    v2f A0, A1;
    A0.x = half ? c2 : 1.0f;  A0.y = half ? c3 : ct;
    A1.x = half ? c6 : c4;    A1.y = half ? 0.0f : c5;

    const int kb = half ? 2 : 0;
    v2f B0, B1;
    B0.x = BMAT[kb][m];      B0.y = BMAT[kb + 1][m];
    B1.x = BMAT[kb + 4][m];  B1.y = BMAT[kb + 5][m];

    v8f C = {0.f, 0.f, 0.f, 0.f, 0.f, 0.f, 0.f, 0.f};
    C = __builtin_amdgcn_wmma_f32_16x16x4_f32(false, A0, false, B0,
                                              (short)0, C, false, false);
    C = __builtin_amdgcn_wmma_f32_16x16x4_f32(false, A1, false, B1,
                                              (short)0, C, false, false);

    // Spill sph (columns 0..6 only) to this wave's LDS slice.
    if (m < NUM_L) {
#pragma unroll
        for (int v = 0; v < 8; ++v) sph_s[wv][half * 8 + v][m] = C[v];
    }
    if (!half) id3_s[wv][m] = g;
    // Intra-wave producer->consumer through LDS: DS ops from one wave are
    // processed in order; the compiler inserts the s_wait_dscnt for the loads.

    const int rowsLeft = (nT - base < 16) ? (nT - base) : 16;

    if (rowsLeft == 16) {
        // Fast path: full tile, no guard, fully unrolled. Store address is
        // linear (base*42 + it*32 + lane) -> one VGPR address + immediates.
        float* orow = out + (size_t)base * LR + lane;
        int tl = 0, j = lane;              // element e = it*32 + lane
#pragma unroll
        for (int it = 0; it < 21; ++it) {
            const unsigned gg = (unsigned)id3_s[wv][tl];
            const int l = (j * 171) >> 10;             // j / 6 for j in [0,42)
            const float s = sph_s[wv][tl][l];
            const float r = table[gg * 42u + (unsigned)j];   // u32 offset
            __builtin_nontemporal_store(r * s, orow + 32 * it);
            j += 32;
            if (j >= LR) { j -= LR; ++tl; }
        }
    } else {
        // Tail tile: guarded, rolled.
        int tl = 0, j = lane;
#pragma unroll 1
        for (int it = 0; it < 21; ++it) {
            if (tl < rowsLeft) {
                const unsigned gg = (unsigned)id3_s[wv][tl];
                const int l = (j * 171) >> 10;
                const float s = sph_s[wv][tl][l];
                const float r = table[gg * 42u + (unsigned)j];
                __builtin_nontemporal_store(
                    r * s, &out[(unsigned)(base + tl) * 42u + (unsigned)j]);
            }
            j += 32;
            if (j >= LR) { j -= LR; ++tl; }
        }
    }
}

// ---------------------------------------------------------------------------
// Fallback (ws too small): fully fused, one thread per triplet.
// ---------------------------------------------------------------------------
__global__ __launch_bounds__(256)
void sbl_fused_kernel(const float* __restrict__ D,
                      const float* __restrict__ angle,
                      const int* __restrict__ id3,
                      float* __restrict__ out, int nT, BesselConsts K) {
    int t = blockIdx.x * blockDim.x + threadIdx.x;
    if (t >= nT) return;
    const int g = id3[t];
    float d = D[g] * 0.2f;
    float f = envelope_scaled(d);

    const float ct = __cosf(angle[t]);
    float sph[NUM_L];
    {   // reference Legendre recurrence
        float Pm = 1.0f, Pc = ct;
        sph[0] = YNORM[0];
        sph[1] = YNORM[1] * ct;
#pragma unroll
        for (int l = 1; l < NUM_L - 1; ++l) {
            float Pn = ((2.0f * l + 1.0f) * ct * Pc - (float)l * Pm) /
                       (float)(l + 1);
            Pm = Pc; Pc = Pn;
            sph[l + 1] = YNORM[l + 1] * Pn;
        }
    }
    float* row = out + (size_t)t * LR;
    rbf_row(d, f, K, [&](int l, int i, float v) {
        __builtin_nontemporal_store(v * sph[l], &row[i]);
    });
}

// ---------------------------------------------------------------------------
// Host: Bessel zeros + norms, exact port of the reference's bisection (f64).
// Deterministic, recomputed every call (no cached state).
// ---------------------------------------------------------------------------
static double sph_jn_h(double x, int n) {
    double j0 = sin(x) / x;
    if (n == 0) return j0;
    double j1 = sin(x) / (x * x) - cos(x) / x;
    double jm1 = j0, j = j1;
    for (int l = 1; l < n; ++l) {
        double nx = (2.0 * l + 1.0) / x * j - jm1;
        jm1 = j; j = nx;
    }
    return j;
}

static double bisect_h(int order, double a, double b) {
    double fa = sph_jn_h(a, order);
    for (int it = 0; it < 80; ++it) {
        double m = 0.5 * (a + b);
        double fm = sph_jn_h(m, order);
        if (fa * fm <= 0.0) { b = m; }
        else { a = m; fa = fm; }
    }
    return 0.5 * (a + b);
}

static void compute_bessel_consts(BesselConsts* K) {
    const double PI = 3.14159265358979323846;
    double Z[NUM_L][NUM_R];
    for (int k = 0; k < NUM_R; ++k) Z[0][k] = (k + 1) * PI;
    double points[NUM_R + NUM_L];      // 12 initial points: (1..12)*pi
    for (int i = 0; i < NUM_R + NUM_L - 1; ++i) points[i] = (i + 1) * PI;
    for (int i = 1; i < NUM_L; ++i) {
        int m = NUM_R + NUM_L - 1 - i;
        double rac[NUM_R + NUM_L];
        for (int j = 0; j < m; ++j)
            rac[j] = bisect_h(i, points[j], points[j + 1]);
        for (int j = 0; j < m; ++j) points[j] = rac[j];
        for (int k = 0; k < NUM_R; ++k) Z[i][k] = rac[k];
    }
    for (int l = 0; l < NUM_L; ++l)
        for (int n = 0; n < NUM_R; ++n) {
            double jv = sph_jn_h(Z[l][n], l + 1);
            K->z[l * NUM_R + n]  = (float)Z[l][n];
            K->bn[l * NUM_R + n] = (float)(1.0 / sqrt(0.5 * jv * jv));
        }
}

// ---------------------------------------------------------------------------
extern "C" void kernel_launch(void* const* d_in, const int* in_sizes, int n_in,
                              void* d_out, int out_size, void* d_ws,
                              size_t ws_size, hipStream_t stream) {
    const float* D     = (const float*)d_in[0];
    const float* angle = (const float*)d_in[1];
    const int*   id3   = (const int*)d_in[2];
    // d_in[3] (Kidx) is unused by the reference.
    float* out = (float*)d_out;
    const int nE = in_sizes[0];
    const int nT = in_sizes[1];

    BesselConsts K;
    compute_bessel_consts(&K);

    const size_t tableBytes = (size_t)nE * LR * sizeof(float);
    if (ws_size >= tableBytes) {
        float* table = (float*)d_ws;
        int blocks1 = (nE + 255) / 256;
        sbl_edge_rbf_kernel<<<blocks1, 256, 0, stream>>>(D, table, nE, K);

        int nTiles  = (nT + 15) / 16;
        int blocks2 = (nTiles + WPB - 1) / WPB;
        sbl_triplet_wmma_kernel<<<blocks2, 256, 0, stream>>>(angle, id3, table,
                                                             out, nT);
    } else {
        int blocks = (nT + 255) / 256;
        sbl_fused_kernel<<<blocks, 256, 0, stream>>>(D, angle, id3, out, nT, K);
    }
}